// SelfAttention_48765058678988
// MI455X (gfx1250) — compile-verified
//
#include <hip/hip_runtime.h>
#include <hip/hip_bf16.h>

// Self-attention with per-batch projection weights, fused via CDNA5 WMMA f32.
// B=32, D=128, L=512, NH=8, DK=DV=16. All GEMMs use V_WMMA_F32_16X16X4_F32
// (full f32 precision; 16x16 tiles match head_dim exactly, wave32-native).
// All unique data (~24 MB) is L2-resident on MI455X (192 MB L2), so the
// optimization targets are WMMA density per memory op and LDS reuse.

#define LSEQ 512
#define DMODEL 128
#define NHEADS 8
#define NBATCH 32

#define QPITCH 520   // 512+8: half-wave A-rows land on disjoint LDS banks
#define VTPITCH 17   // transposed V rows padded: conflict-free lane-consecutive reads

typedef float v2f __attribute__((ext_vector_type(2)));
typedef float v8f __attribute__((ext_vector_type(8)));

__device__ __forceinline__ v8f wmma4(v2f a, v2f b, v8f c) {
    // (neg_a, A, neg_b, B, c_mod, C, reuse_a, reuse_b)
    return __builtin_amdgcn_wmma_f32_16x16x4_f32(false, a, false, b, (short)0, c,
                                                 false, false);
}

// A fragment (16x4, MxK): element (m,k) at p[m*ld + k].
// Lane layout: m = lane&15, k = 2*(lane>>4) + vgpr.
__device__ __forceinline__ v2f load_a_rm(const float* p, int ld) {
    int lane = threadIdx.x & 31;
    int m = lane & 15, k = (lane >> 4) * 2;
    v2f r;
    r.x = p[m * ld + k];
    r.y = p[m * ld + k + 1];
    return r;
}

// A fragment where logical A is the transpose of storage: (m,k) at p[k*ld + m].
__device__ __forceinline__ v2f load_a_cm(const float* p, int ld) {
    int lane = threadIdx.x & 31;
    int m = lane & 15, k = (lane >> 4) * 2;
    v2f r;
    r.x = p[k * ld + m];
    r.y = p[(k + 1) * ld + m];
    return r;
}

// B fragment (4x16, KxN): element (k,n) at p[k*ld + n].
__device__ __forceinline__ v2f load_b_rm(const float* p, int ld) {
    int lane = threadIdx.x & 31;
    int n = lane & 15, k = (lane >> 4) * 2;
    v2f r;
    r.x = p[k * ld + n];
    r.y = p[(k + 1) * ld + n];
    return r;
}

// C/D 16x16 f32 layout: lane<16 -> (M=r, N=lane); lane>=16 -> (M=r+8, N=lane-16).
__device__ __forceinline__ void store_c(float* p, int ld, v8f c) {
    int lane = threadIdx.x & 31;
    int col = lane & 15, rbase = (lane >> 4) * 8;
#pragma unroll
    for (int r = 0; r < 8; ++r) p[(size_t)(rbase + r) * ld + col] = c[r];
}

// ---------------------------------------------------------------------------
// Kernel 1: Q/K/V projection. One block per (head,batch); 8 waves.
// Each x B-fragment is loaded once and feeds three WMMAs (Q,K,V accumulators).
// ---------------------------------------------------------------------------
__global__ __launch_bounds__(256) void qkv_kernel(const float* __restrict__ x,
                                                  const float* __restrict__ Wqs,
                                                  const float* __restrict__ Wks,
                                                  const float* __restrict__ Wvs,
                                                  float* __restrict__ Q,
                                                  float* __restrict__ K,
                                                  float* __restrict__ V) {
    int wave = threadIdx.x >> 5;
    int hb = blockIdx.x;          // n*NBATCH + b
    int b = hb & (NBATCH - 1);
    const float* xb = x + (size_t)b * DMODEL * LSEQ;
    const float* Wq = Wqs + (size_t)hb * 16 * DMODEL;
    const float* Wk = Wks + (size_t)hb * 16 * DMODEL;
    const float* Wv = Wvs + (size_t)hb * 16 * DMODEL;
    float* Qo = Q + (size_t)hb * 16 * LSEQ;
    float* Ko = K + (size_t)hb * 16 * LSEQ;
    float* Vo = V + (size_t)hb * 16 * LSEQ;

#pragma unroll
    for (int j = 0; j < 4; ++j) {          // 4 column tiles per wave
        int n0 = (wave + 8 * j) * 16;
        v8f aq = {}, ak = {}, av = {};
#pragma unroll 4
        for (int k0 = 0; k0 < DMODEL; k0 += 4) {
            v2f bb = load_b_rm(xb + (size_t)k0 * LSEQ + n0, LSEQ);  // shared B
            aq = wmma4(load_a_rm(Wq + k0, DMODEL), bb, aq);
            ak = wmma4(load_a_rm(Wk + k0, DMODEL), bb, ak);
            av = wmma4(load_a_rm(Wv + k0, DMODEL), bb, av);
        }
        store_c(Qo + n0, LSEQ, aq);
        store_c(Ko + n0, LSEQ, ak);
        store_c(Vo + n0, LSEQ, av);
    }
}

// ---------------------------------------------------------------------------
// Kernel 2: attention with online softmax over the l axis (per column m).
// scores[l,m] = sum_k Q[k,l]*K[k,m] * 0.25 ; S = softmax_l ; head = V @ S.
// One wave per 16-wide m-tile; 4 waves per block share (head,batch), so Q and
// V are staged in LDS once per block. K fragments are l-invariant -> hoisted.
// ---------------------------------------------------------------------------
__global__ __launch_bounds__(128) void attn_kernel(const float* __restrict__ Q,
                                                   const float* __restrict__ K,
                                                   const float* __restrict__ V,
                                                   float* __restrict__ H) {
    int wave = threadIdx.x >> 5;
    int lane = threadIdx.x & 31;
    int tid = threadIdx.x;
    int hb = blockIdx.x >> 3;               // (head,batch)
    int mgroup = blockIdx.x & 7;
    int n = hb >> 5;                        // head
    int b = hb & (NBATCH - 1);
    int m0 = (mgroup * 4 + wave) * 16;

    const float* Qp = Q + (size_t)hb * 16 * LSEQ;
    const float* Kp = K + (size_t)hb * 16 * LSEQ;
    const float* Vp = V + (size_t)hb * 16 * LSEQ;

    __shared__ float Qs[16 * QPITCH];       // Q row-major, padded pitch
    __shared__ float Vt[LSEQ * VTPITCH];    // V transposed: Vt[l][v]
    __shared__ float Pl[4][16 * 17];        // per-wave P tile, padded

    // Stage Q (vectorized, coalesced) and V (transposed) into LDS.
    for (int i = tid * 4; i < 16 * LSEQ; i += 128 * 4) {
        int r = i >> 9, c = i & (LSEQ - 1);
        *(float4*)&Qs[r * QPITCH + c] = *(const float4*)&Qp[i];
    }
    for (int i = tid; i < 16 * LSEQ; i += 128) {
        int v = i >> 9, l = i & (LSEQ - 1);
        Vt[l * VTPITCH + v] = Vp[i];
    }
    // K B-fragments for this wave's m-tile are l-invariant: load once.
    v2f bK[4];
#pragma unroll
    for (int kk = 0; kk < 4; ++kk)
        bK[kk] = load_b_rm(Kp + (size_t)(4 * kk) * LSEQ + m0, LSEQ);
    __syncthreads();

    float* myP = Pl[wave];
    int col = lane & 15, rbase = (lane >> 4) * 8;
    float rmax = -__builtin_inff();
    float rsum = 0.0f;
    v8f acc = {};
    const float scale = 0.25f;              // 1/sqrt(DK), DK=16

    for (int l0 = 0; l0 < LSEQ; l0 += 16) {
        // S_tile[l,m] = sum_k Q[k, l0+l] * K[k, m0+m]
        v8f s = {};
#pragma unroll
        for (int kk = 0; kk < 4; ++kk) {
            v2f a = load_a_cm(Qs + (size_t)(4 * kk) * QPITCH + l0, QPITCH);
            s = wmma4(a, bK[kk], s);
        }

        // online softmax: each lane owns column m0+col, rows rbase..rbase+7
        float lm = -__builtin_inff();
#pragma unroll
        for (int r = 0; r < 8; ++r) {
            s[r] *= scale;
            lm = fmaxf(lm, s[r]);
        }
        float tm = fmaxf(lm, __shfl_xor(lm, 16, 32));  // combine both halves
        float newmax = fmaxf(rmax, tm);
        float factor = __expf(rmax - newmax);
        rmax = newmax;

        float p[8], lsum = 0.0f;
#pragma unroll
        for (int r = 0; r < 8; ++r) {
            p[r] = __expf(s[r] - newmax);
            lsum += p[r];
        }
        float tsum = lsum + __shfl_xor(lsum, 16, 32);
        rsum = rsum * factor + tsum;
#pragma unroll
        for (int r = 0; r < 8; ++r) acc[r] *= factor;

        // C-layout -> B-layout via LDS
#pragma unroll
        for (int r = 0; r < 8; ++r) myP[(rbase + r) * 17 + col] = p[r];
        __syncthreads();

        // acc[v,m] += sum_l V[v, l0+l] * P[l, m]; A read from transposed Vt
#pragma unroll
        for (int kk = 0; kk < 4; ++kk) {
            const float* vt = Vt + (size_t)(l0 + 4 * kk) * VTPITCH;
            int k2 = (lane >> 4) * 2;
            v2f a;
            a.x = vt[k2 * VTPITCH + col];          // V[col][l0+4kk+k2]
            a.y = vt[(k2 + 1) * VTPITCH + col];
            v2f bb = load_b_rm(myP + 4 * kk * 17, 17);
            acc = wmma4(a, bb, acc);
        }
        __syncthreads();
    }

    // normalize and store with reversed head order: row block = (NH-1-n)
    float inv = 1.0f / rsum;
    float* Hb = H + (size_t)b * DMODEL * LSEQ;
    int rowblk = (NHEADS - 1 - n) * 16;
#pragma unroll
    for (int r = 0; r < 8; ++r)
        Hb[(size_t)(rowblk + rbase + r) * LSEQ + m0 + col] = acc[r] * inv;
}

// ---------------------------------------------------------------------------
// Kernel 3: out[b] = Wo[b](128x128) @ H[b](128x512). Each wave owns one
// 16-row block and 4 column tiles: every Wo A-fragment feeds 4 WMMAs.
// ---------------------------------------------------------------------------
__global__ __launch_bounds__(256) void out_kernel(const float* __restrict__ Wo,
                                                  const float* __restrict__ H,
                                                  float* __restrict__ out) {
    int wave = threadIdx.x >> 5;
    int b = blockIdx.x >> 3;
    int n0base = (blockIdx.x & 7) * 64;        // 4 tiles of 16 columns
    int m0 = wave * 16;
    const float* Wb = Wo + (size_t)b * DMODEL * DMODEL;
    const float* Hb = H + (size_t)b * DMODEL * LSEQ;
    float* Ob = out + (size_t)b * DMODEL * LSEQ;

    v8f acc0 = {}, acc1 = {}, acc2 = {}, acc3 = {};
#pragma unroll 4
    for (int k0 = 0; k0 < DMODEL; k0 += 4) {
        v2f a = load_a_rm(Wb + (size_t)m0 * DMODEL + k0, DMODEL);  // shared A
        const float* hrow = Hb + (size_t)k0 * LSEQ + n0base;
        acc0 = wmma4(a, load_b_rm(hrow, LSEQ), acc0);
        acc1 = wmma4(a, load_b_rm(hrow + 16, LSEQ), acc1);
        acc2 = wmma4(a, load_b_rm(hrow + 32, LSEQ), acc2);
        acc3 = wmma4(a, load_b_rm(hrow + 48, LSEQ), acc3);
    }
    float* obase = Ob + (size_t)m0 * LSEQ + n0base;
    store_c(obase, LSEQ, acc0);
    store_c(obase + 16, LSEQ, acc1);
    store_c(obase + 32, LSEQ, acc2);
    store_c(obase + 48, LSEQ, acc3);
}

extern "C" void kernel_launch(void* const* d_in, const int* in_sizes, int n_in,
                              void* d_out, int out_size, void* d_ws, size_t ws_size,
                              hipStream_t stream) {
    const float* x = (const float*)d_in[0];
    const float* Wqs = (const float*)d_in[1];
    const float* Wks = (const float*)d_in[2];
    const float* Wvs = (const float*)d_in[3];
    const float* Wo = (const float*)d_in[4];

    // Workspace: Q,K,V (NH*B*16*L floats each) + H (B*128*L floats) = 32 MB
    const size_t seg = (size_t)NHEADS * NBATCH * 16 * LSEQ;  // 2,097,152 floats
    float* ws = (float*)d_ws;
    float* Q = ws;
    float* K = ws + seg;
    float* V = ws + 2 * seg;
    float* H = ws + 3 * seg;

    qkv_kernel<<<NHEADS * NBATCH, 256, 0, stream>>>(x, Wqs, Wks, Wvs, Q, K, V);
    attn_kernel<<<NHEADS * NBATCH * 8, 128, 0, stream>>>(Q, K, V, H);
    out_kernel<<<NBATCH * 8, 256, 0, stream>>>(Wo, H, (float*)d_out);
}